// DGS3DLayer_83726092468929
// MI455X (gfx1250) — compile-verified
//
#include <hip/hip_runtime.h>

// CDNA5 / gfx1250: wave32, WMMA 16x16 shapes.
typedef __attribute__((ext_vector_type(2))) float v2f;
typedef __attribute__((ext_vector_type(8))) float v8f;

#define DD 128
#define HH 128
#define WW 128
#define CC 16
#define DHW (DD * HH * WW)

// -------- Pass 1: (b,c,D,H,W) -> (b,D,H,W,c). float4-vectorized both directions.
// Each thread handles 4 consecutive spatial positions: 16x b128 loads, 16x b128 stores.
__global__ __launch_bounds__(256) void dgs3d_transpose_kernel(
    const float* __restrict__ in, float* __restrict__ out, int totalVec)
{
    int t = blockIdx.x * blockDim.x + threadIdx.x;
    if (t >= totalVec) return;
    const int VECS = DHW / 4;
    int bi  = t / VECS;
    int spv = t - bi * VECS;

    const float* src = in + (size_t)bi * CC * DHW + (size_t)spv * 4;
    float vr[4][CC];                       // [position j][channel]
#pragma unroll
    for (int ch = 0; ch < CC; ++ch) {
        float4 ld = *(const float4*)(src + (size_t)ch * DHW);
        vr[0][ch] = ld.x; vr[1][ch] = ld.y; vr[2][ch] = ld.z; vr[3][ch] = ld.w;
    }
    float* dstb = out + ((size_t)bi * DHW + (size_t)spv * 4) * CC;  // 256B/lane contiguous
#pragma unroll
    for (int j = 0; j < 4; ++j) {
        float4* dst = (float4*)(dstb + j * CC);
#pragma unroll
        for (int r = 0; r < 4; ++r)
            dst[r] = make_float4(vr[j][4 * r + 0], vr[j][4 * r + 1],
                                 vr[j][4 * r + 2], vr[j][4 * r + 3]);
    }
}

// -------- Pass 2: one wave per query.
// D(16ch x 4out) = A(16ch x 8corners) * B(8corners x 4weights), two accumulating
// V_WMMA_F32_16X16X4_F32 (corners split by z-plane). Strides are template constants
// so gather offsets stay in 32-bit VALU with an SGPR 64-bit base.
template <int SZ, int SY, int SX, int SC>
__global__ __launch_bounds__(256) void dgs3d_sample_wmma_kernel(
    const float* __restrict__ vol, const float* __restrict__ grid,
    const float* __restrict__ psx, const float* __restrict__ psy, const float* __restrict__ psz,
    float* __restrict__ out, int q, int nb)
{
    const int lane = threadIdx.x & 31;
    const int half = lane >> 4;   // K-half for A/B rows, M-half for D
    const int m    = lane & 15;   // A: channel (M);  B/D: column (N)
    const int wave   = (int)((blockIdx.x * blockDim.x + threadIdx.x) >> 5);
    const int nwaves = (int)((gridDim.x * blockDim.x) >> 5);

    const float scX = (float)(WW - 1) / psx[0];
    const float scY = (float)(HH - 1) / psy[0];
    const float scZ = (float)(DD - 1) / psz[0];

    // Loop-invariant per-lane weight selectors (branchless cndmask sources).
    const bool  is1 = (m == 1), is2 = (m == 2), is3 = (m == 3), valid = (m < 4);
    const float syscY = half ? scY : -scY;   // dy sign * gy scale
    const float fxs   = is1 ? 1.f : 0.f;     // unused placeholder-free form below

    for (int bi = 0; bi < nb; ++bi) {        // wave-uniform
        const float* volb = vol + (size_t)bi * (CC * (size_t)DHW);
        const float* gb   = grid + (size_t)bi * q * 3;
        float*       ob   = out + (size_t)bi * CC * 4 * (size_t)q;

        for (int qi = wave; qi < q; qi += nwaves) {   // wave-uniform: EXEC all-1s at WMMA
            const float* g = gb + (size_t)qi * 3;
            __builtin_prefetch(gb + (size_t)(qi + nwaves) * 3, 0, 1);  // global_prefetch_b8
            const float gx = g[0], gy = g[1], gz = g[2];

            float x = fminf(fmaxf((gx + 1.f) * (0.5f * (WW - 1)), 0.f), (float)(WW - 1));
            float y = fminf(fmaxf((gy + 1.f) * (0.5f * (HH - 1)), 0.f), (float)(HH - 1));
            float z = fminf(fmaxf((gz + 1.f) * (0.5f * (DD - 1)), 0.f), (float)(DD - 1));
            int x0 = (int)floorf(x); x0 = x0 < 0 ? 0 : (x0 > WW - 2 ? WW - 2 : x0);
            int y0 = (int)floorf(y); y0 = y0 < 0 ? 0 : (y0 > HH - 2 ? HH - 2 : y0);
            int z0 = (int)floorf(z); z0 = z0 < 0 ? 0 : (z0 > DD - 2 ? DD - 2 : z0);
            const float u = x - (float)x0, v = y - (float)y0, w = z - (float)z0;

            // A gathers: lane = channel m, half selects dy; corner k = dz*4 + dy*2 + dx.
            const int off = z0 * SZ + (y0 + half) * SY + x0 * SX + m * SC;
            v2f a1, a2;
            a1.x = volb[off];            // (dz=0, dy=half, dx=0) -> K = 0|2
            a1.y = volb[off + SX];       // (dz=0, dy=half, dx=1) -> K = 1|3
            a2.x = volb[off + SZ];       // (dz=1, ...)           -> K = 4|6
            a2.y = volb[off + SZ + SX];  //                        -> K = 5|7

            // Branchless B weights. Column n picks which factor flips to a derivative.
            const float wyh = half ? v : (1.f - v);
            const float fy  = valid ? (is2 ? syscY : wyh) : 0.f;
            const float fx0 = is1 ? -scX : (1.f - u);
            const float fx1 = is1 ?  scX : u;
            const float fz0 = is3 ? -scZ : (1.f - w);
            const float fz1 = is3 ?  scZ : w;
            v2f b1, b2;
            const float fz0y = fz0 * fy, fz1y = fz1 * fy;
            b1.x = fz0y * fx0;  b1.y = fz0y * fx1;   // dz=0, dx=0|1
            b2.x = fz1y * fx0;  b2.y = fz1y * fx1;   // dz=1, dx=0|1

            v8f acc = {};
            acc = __builtin_amdgcn_wmma_f32_16x16x4_f32(false, a1, false, b1,
                                                        (short)0, acc, false, false);
            acc = __builtin_amdgcn_wmma_f32_16x16x4_f32(false, a2, false, b2,
                                                        (short)0, acc, false, false);

            // D: VGPR r -> channel M = r + 8*half, column N = m. Pointer-increment stores.
            if (valid) {
                float* op = ob + ((size_t)((8 * half) * 4 + m)) * q + qi;
                const size_t rstride = 4 * (size_t)q;   // between channels in (b,c,4,q)
#pragma unroll
                for (int r = 0; r < 8; ++r) {
                    *op = acc[r];
                    op += rstride;
                }
            }
        }
    }
}

extern "C" void kernel_launch(void* const* d_in, const int* in_sizes, int n_in,
                              void* d_out, int out_size, void* d_ws, size_t ws_size,
                              hipStream_t stream)
{
    const float* input = (const float*)d_in[0];
    const float* grid  = (const float*)d_in[1];
    const float* psx   = (const float*)d_in[2];
    const float* psy   = (const float*)d_in[3];
    const float* psz   = (const float*)d_in[4];
    float* out = (float*)d_out;

    const int b = in_sizes[0] / (CC * DHW);
    const int q = in_sizes[1] / (b * 3);

    const size_t volBytes = (size_t)b * DHW * CC * sizeof(float);
    const int sampleBlocks = 2048;   // 16384 waves over b*q = 200k query-tasks

    if (ws_size >= volBytes) {
        // Fast path: channel-contiguous staging volume in workspace.
        float* vws = (float*)d_ws;
        const int totalVec = b * (DHW / 4);
        dgs3d_transpose_kernel<<<(totalVec + 255) / 256, 256, 0, stream>>>(input, vws, totalVec);
        dgs3d_sample_wmma_kernel<HH * WW * CC, WW * CC, CC, 1>
            <<<sampleBlocks, 256, 0, stream>>>(vws, grid, psx, psy, psz, out, q, b);
    } else {
        // Fallback: gather straight from (b,c,D,H,W) with strided channels.
        dgs3d_sample_wmma_kernel<HH * WW, WW, 1, DHW>
            <<<sampleBlocks, 256, 0, stream>>>(input, grid, psx, psy, psz, out, q, b);
    }
}